// ElementalLinear_74225624809590
// MI455X (gfx1250) — compile-verified
//
#include <hip/hip_runtime.h>

// Problem constants (from reference)
#define N_ATOMS  50000
#define NUM_ELEM 10
#define LMAX     3
#define SH_DIM   16        // (LMAX+1)^2
#define C_IN     128
#define C_OUT    128
#define TILE_M   16
#define NUM_TILES ((N_ATOMS + TILE_M - 1) / TILE_M)   // 3125

typedef float v2f __attribute__((ext_vector_type(2)));
typedef float v8f __attribute__((ext_vector_type(8)));

// ---------------------------------------------------------------------------
// Pass 0: zero species counters (graph-capture-safe, no hipMemset needed)
// ---------------------------------------------------------------------------
__global__ void zero_counts_kernel(int* __restrict__ counts) {
    if (threadIdx.x < NUM_ELEM) counts[threadIdx.x] = 0;
}

// ---------------------------------------------------------------------------
// Pass 1: bucket atoms by species (argmax of one-hot y). Order within a
// bucket is nondeterministic but output values are independent of it.
// ---------------------------------------------------------------------------
__global__ void bucket_kernel(const float* __restrict__ y,
                              int* __restrict__ counts,
                              int* __restrict__ bucket) {
    int a = blockIdx.x * blockDim.x + threadIdx.x;
    if (a >= N_ATOMS) return;
    const float* yr = y + (long long)a * NUM_ELEM;
    int s = 0;
    float best = yr[0];
#pragma unroll
    for (int e = 1; e < NUM_ELEM; ++e) {
        float v = yr[e];
        if (v > best) { best = v; s = e; }
    }
    int pos = atomicAdd(&counts[s], 1);
    bucket[s * N_ATOMS + pos] = a;
}

// ---------------------------------------------------------------------------
// Pass 2: grouped GEMM via V_WMMA_F32_16X16X4_F32.
// Block = 16 atoms of one species. 8 waves; wave w owns C_OUT cols [16w,16w+16).
// For each l: hoist B-tile (128x16 slice of W[s,l]) into regs, then for each
// of the (2l+1) SH rows do a 16x16x128 WMMA chain (32 x K=4 steps).
// ---------------------------------------------------------------------------
__global__ void __launch_bounds__(256)
elemental_linear_wmma(const float* __restrict__ x,
                      const float* __restrict__ w,
                      float* __restrict__ out,
                      const int* __restrict__ counts,
                      const int* __restrict__ bucket) {
    __shared__ int s_atoms[TILE_M];
    __shared__ int s_nvalid;

    const int sp   = blockIdx.y;             // species
    const int tile = blockIdx.x;             // atom tile within species
    const int cnt  = counts[sp];             // block-uniform
    const int base = tile * TILE_M;
    if (base >= cnt) return;                 // uniform exit: EXEC stays full

    if (threadIdx.x < TILE_M) {
        int idx  = base + threadIdx.x;
        int last = cnt - 1;
        if (idx > last) idx = last;          // clamp: safe duplicate reads
        s_atoms[threadIdx.x] = bucket[sp * N_ATOMS + idx];
    }
    if (threadIdx.x == 0) s_nvalid = min(TILE_M, cnt - base);
    __syncthreads();

    const int lane = threadIdx.x & 31;
    const int wv   = threadIdx.x >> 5;       // 0..7 == N tile index
    const int half = lane >> 4;              // 0|1 : K sub-row pair select
    const int nn   = lane & 15;              // N (B/C/D) or M (A) per layout
    const int ncol = wv * 16 + nn;
    const int nvalid = s_nvalid;

    // A fragment source row: lane nn owns atom nn (both lane halves)
    const long long arow_base = (long long)s_atoms[nn] * (SH_DIM * C_IN);
    const float scale = 0.08838834764831845f;   // 1/sqrt(128)

    for (int l = 0; l <= LMAX; ++l) {
        // --- hoist B tile: W[sp][l][0:128, ncol], laid out per WMMA B spec:
        // step k VGPR0 = K row 4k+2*half, VGPR1 = K row 4k+1+2*half
        const float* wl = w + ((long long)(sp * (LMAX + 1) + l) * C_IN) * C_OUT;
        v2f bfrag[32];
#pragma unroll
        for (int k = 0; k < 32; ++k) {
            int kr = 4 * k + 2 * half;
            bfrag[k].x = wl[(long long)kr * C_OUT + ncol];
            bfrag[k].y = wl[(long long)(kr + 1) * C_OUT + ncol];
        }

        const int r0 = l * l;
        const int nr = 2 * l + 1;
        for (int rr = 0; rr < nr; ++rr) {
            const int r = r0 + rr;
            const float* xr = x + arow_base + (long long)r * C_IN;

            // A fragments: contiguous float2 (global_load_b64), kr is even
            v2f afrag[32];
#pragma unroll
            for (int k = 0; k < 32; ++k) {
                int kr = 4 * k + 2 * half;
                afrag[k] = *(const v2f*)(xr + kr);
            }

            v8f c = {};
#pragma unroll
            for (int k = 0; k < 32; ++k) {
                c = __builtin_amdgcn_wmma_f32_16x16x4_f32(
                        /*neg_a=*/false, afrag[k],
                        /*neg_b=*/false, bfrag[k],
                        /*c_mod=*/(short)0, c,
                        /*reuse_a=*/false, /*reuse_b=*/false);
            }

            // C/D layout: VGPR v, lanes 0-15 -> M=v, lanes 16-31 -> M=v+8
#pragma unroll
            for (int v = 0; v < 8; ++v) {
                int row = v + 8 * half;
                if (row < nvalid) {
                    long long o = (long long)s_atoms[row] * (SH_DIM * C_OUT)
                                + (long long)r * C_OUT + ncol;
                    out[o] = c[v] * scale;
                }
            }
        }
    }
}

// ---------------------------------------------------------------------------
extern "C" void kernel_launch(void* const* d_in, const int* in_sizes, int n_in,
                              void* d_out, int out_size, void* d_ws, size_t ws_size,
                              hipStream_t stream) {
    const float* x = (const float*)d_in[0];   // (50000,16,128) f32
    const float* y = (const float*)d_in[1];   // (50000,10)     f32 one-hot
    const float* w = (const float*)d_in[2];   // (10,4,128,128) f32
    float* out = (float*)d_out;               // (50000,16,128) f32

    // workspace: [0..9] counts (padded to 16 ints), then 10*N_ATOMS bucket ids
    int* counts = (int*)d_ws;
    int* bucket = counts + 16;

    zero_counts_kernel<<<1, 32, 0, stream>>>(counts);
    bucket_kernel<<<(N_ATOMS + 255) / 256, 256, 0, stream>>>(y, counts, bucket);

    dim3 grid(NUM_TILES, NUM_ELEM);
    elemental_linear_wmma<<<grid, 256, 0, stream>>>(x, w, out, counts, bucket);
}